// DenoisingConvNN_2D_K_N_25039659335747
// MI455X (gfx1250) — compile-verified
//
#include <hip/hip_runtime.h>
#include <math.h>

typedef __attribute__((ext_vector_type(2))) float v2f;
typedef __attribute__((ext_vector_type(8))) float v8f;

#define TT   4096   // T = (128/2)*(128/2)
#define NS   64     // number of sampled columns
#define KNBR 9      // top-k
#define NB   32     // batch

// ---------------- pixel_unshuffle: x(B,3,128,128) -> t1(B,12,4096) ----------
__global__ __launch_bounds__(256) void k_unshuffle(const float* __restrict__ x,
                                                   float* __restrict__ t1) {
  int i = blockIdx.x * blockDim.x + threadIdx.x;
  if (i >= NB * 12 * TT) return;
  int tt = i & 4095;
  int cp = (i >> 12) % 12;
  int b  = i / (12 * TT);
  int c = cp >> 2, s1 = (cp >> 1) & 1, s2 = cp & 1;
  int hh = tt >> 6, ww = tt & 63;
  t1[i] = x[((size_t)(b * 3 + c) * 128 + (2 * hh + s1)) * 128 + (2 * ww + s2)];
}

// ---------------- pixel_shuffle: y3(B,12,4096) -> out(B,3,128,128) ----------
__global__ __launch_bounds__(256) void k_shuffle(const float* __restrict__ y3,
                                                 float* __restrict__ out) {
  int i = blockIdx.x * blockDim.x + threadIdx.x;
  if (i >= NB * 3 * 128 * 128) return;
  int w = i & 127, h = (i >> 7) & 127;
  int c = (i >> 14) % 3, b = i / (3 * 16384);
  int s1 = h & 1, s2 = w & 1, hh = h >> 1, ww = w >> 1;
  out[i] = y3[((size_t)b * 12 + (c * 4 + s1 * 2 + s2)) * TT + hh * 64 + ww];
}

// ------- gather sample columns + column-normalize (per batch block) ---------
__global__ __launch_bounds__(64) void k_sample(const float* __restrict__ t,
                                               const int* __restrict__ idx,
                                               float* __restrict__ samp,
                                               float* __restrict__ sn, int C) {
  int b = blockIdx.x, n = threadIdx.x;            // n in [0,64)
  int j = idx[n];
  const float* tb = t + (size_t)b * C * TT;
  float* sb  = samp + (size_t)b * C * NS;
  float* snb = sn   + (size_t)b * C * NS;
  float acc = 0.f;
  for (int c = 0; c < C; ++c) {
    float v = tb[(size_t)c * TT + j];
    sb[c * NS + n] = v;
    acc += v * v;
  }
  float inv = 1.0f / fmaxf(sqrtf(acc), 1e-12f);
  for (int c = 0; c < C; ++c) snb[c * NS + n] = sb[c * NS + n] * inv;
}

// ---- sim = t^T * sn via V_WMMA_F32_16X16X4_F32, fused stable top-9 ---------
// block = 128 threads (4 waves); each wave: one 16-row M-tile over N=64.
// Row scale 1/||t_col|| is a positive per-row constant -> skipped (ordering-
// invariant; sim values are never emitted, only nbr indices).
__global__ __launch_bounds__(128) void k_sim_topk(const float* __restrict__ t,
                                                  const float* __restrict__ sn,
                                                  int* __restrict__ nbr, int C) {
  __shared__ float simLDS[64 * 68];  // stride 68 to dodge bank conflicts
  int b = blockIdx.y;
  int t0 = blockIdx.x * 64;
  int tid = threadIdx.x;
  int lane = tid & 31, wave = tid >> 5;
  const float* tb  = t  + (size_t)b * C * TT;
  const float* snb = sn + (size_t)b * C * NS;

  v8f zero = {};
  v8f acc[4] = {zero, zero, zero, zero};
  int m  = lane & 15;
  int kb = (lane >> 4) * 2;
  int trow = t0 + wave * 16 + m;
  for (int c0 = 0; c0 < C; c0 += 4) {
    v2f a;
    a.x = tb[(size_t)(c0 + kb)     * TT + trow];
    a.y = tb[(size_t)(c0 + kb + 1) * TT + trow];
#pragma unroll
    for (int j = 0; j < 4; ++j) {
      v2f bf;
      bf.x = snb[(c0 + kb)     * NS + j * 16 + m];
      bf.y = snb[(c0 + kb + 1) * NS + j * 16 + m];
      acc[j] = __builtin_amdgcn_wmma_f32_16x16x4_f32(
          false, a, false, bf, (short)0, acc[j], false, false);
    }
  }
#pragma unroll
  for (int j = 0; j < 4; ++j)
#pragma unroll
    for (int r = 0; r < 8; ++r) {
      int row = wave * 16 + (lane >> 4) * 8 + r;     // C/D layout
      simLDS[row * 68 + j * 16 + m] = acc[j][r];
    }
  __syncthreads();

  if (tid < 64) {                                    // one thread per t-row
    float bv[KNBR]; int bi[KNBR];
#pragma unroll
    for (int q = 0; q < KNBR; ++q) { bv[q] = -3.4e38f; bi[q] = 0; }
    const float* row = &simLDS[tid * 68];
    for (int n = 0; n < NS; ++n) {                   // stable desc. insertion
      float v = row[n];
      if (v > bv[KNBR - 1]) {
        int p = KNBR - 1;
        while (p > 0 && bv[p - 1] < v) { bv[p] = bv[p - 1]; bi[p] = bi[p - 1]; --p; }
        bv[p] = v; bi[p] = n;
      }
    }
    int* nb = nbr + ((size_t)b * TT + (t0 + tid)) * KNBR;
#pragma unroll
    for (int q = 0; q < KNBR; ++q) nb[q] = bi[q];
  }
}

// ---- y[o,t] = sum_{k,c} w[o,c,k]*samp[c,nbr[t,k]] + bias, WMMA f32 ---------
// grid (T/64, B); block = 32 * ceil(O/16) threads: wave w owns output m-tile
// w, all waves share ONE LDS samp tile (8x less global samp traffic vs
// per-m-tile blocks on the dominant layer-2 GEMM).
__global__ __launch_bounds__(256) void k_conv(const float* __restrict__ samp,
                                              const int* __restrict__ nbr,
                                              const float* __restrict__ w,
                                              const float* __restrict__ bias,
                                              float* __restrict__ out,
                                              int C, int O, int relu) {
  __shared__ float sampLDS[128 * NS];                // up to 32 KB
  int b  = blockIdx.y;
  int t0 = blockIdx.x * 64;
  int lane = threadIdx.x & 31;
  int wave = threadIdx.x >> 5;
  int m0   = wave * 16;
  const float* sb = samp + (size_t)b * C * NS;
  for (int i = threadIdx.x; i < C * NS; i += blockDim.x) sampLDS[i] = sb[i];
  __syncthreads();

  v8f zero = {};
  v8f acc[4] = {zero, zero, zero, zero};
  int m  = lane & 15;
  int kb = (lane >> 4) * 2;
  int o  = m0 + m;
  float msk = (o < O) ? 1.0f : 0.0f;                 // pad rows -> zero A
  int   oc  = (o < O) ? o : (O - 1);
  const int* nb = nbr + (size_t)b * TT * KNBR;

  for (int ks = 0; ks < KNBR; ++ks) {
    int jn[4];
#pragma unroll
    for (int j = 0; j < 4; ++j) jn[j] = nb[(t0 + j * 16 + m) * KNBR + ks];
    for (int c0 = 0; c0 < C; c0 += 4) {
      v2f a;
      a.x = msk * w[((size_t)oc * C + (c0 + kb))     * KNBR + ks];
      a.y = msk * w[((size_t)oc * C + (c0 + kb + 1)) * KNBR + ks];
#pragma unroll
      for (int j = 0; j < 4; ++j) {
        v2f bf;
        bf.x = sampLDS[(c0 + kb)     * NS + jn[j]];
        bf.y = sampLDS[(c0 + kb + 1) * NS + jn[j]];
        acc[j] = __builtin_amdgcn_wmma_f32_16x16x4_f32(
            false, a, false, bf, (short)0, acc[j], false, false);
      }
    }
  }
#pragma unroll
  for (int j = 0; j < 4; ++j)
#pragma unroll
    for (int r = 0; r < 8; ++r) {
      int M = (lane >> 4) * 8 + r;
      int oo = m0 + M;
      if (oo < O) {
        float v = acc[j][r] + bias[oo];
        if (relu) v = fmaxf(v, 0.0f);
        out[((size_t)b * O + oo) * TT + (t0 + j * 16 + m)] = v;
      }
    }
}

extern "C" void kernel_launch(void* const* d_in, const int* in_sizes, int n_in,
                              void* d_out, int out_size, void* d_ws, size_t ws_size,
                              hipStream_t stream) {
  const float* x    = (const float*)d_in[0];
  const int*   idx1 = (const int*)  d_in[1];
  const int*   idx2 = (const int*)  d_in[2];
  const int*   idx3 = (const int*)  d_in[3];
  const float* w1   = (const float*)d_in[4];
  const float* b1   = (const float*)d_in[5];
  const float* w2   = (const float*)d_in[6];
  const float* b2   = (const float*)d_in[7];
  const float* w3   = (const float*)d_in[8];
  const float* b3   = (const float*)d_in[9];
  float* out = (float*)d_out;

  char* ws = (char*)d_ws;
  float* t1   = (float*)(ws + 0);            // 32*12*4096 f32  (6 MB)
  float* t2   = (float*)(ws + 6291456);      // 32*64*4096 f32  (32 MB)
  float* t3   = (float*)(ws + 39845888);     // 32*128*4096 f32 (64 MB)
  float* samp = (float*)(ws + 106954752);    // 32*128*64 f32   (1 MB)
  float* sn   = (float*)(ws + 108003328);    // 32*128*64 f32   (1 MB)
  int*   nbr  = (int*)  (ws + 109051904);    // 32*4096*9 i32   (4.7 MB)
  float* y3   = t1;                          // t1 dead after layer 1 -> reuse

  k_unshuffle<<<(NB * 12 * TT + 255) / 256, 256, 0, stream>>>(x, t1);

  auto layer = [&](const float* t, const int* idx, const float* w,
                   const float* bias, float* outT, int C, int O, int relu) {
    int mtiles = (O + 15) / 16;
    k_sample  <<<NB, 64, 0, stream>>>(t, idx, samp, sn, C);
    k_sim_topk<<<dim3(TT / 64, NB), 128, 0, stream>>>(t, sn, nbr, C);
    k_conv    <<<dim3(TT / 64, NB), 32 * mtiles, 0, stream>>>(
        samp, nbr, w, bias, outT, C, O, relu);
  };

  layer(t1, idx1, w1, b1, t2, 12,  64, 1);   // relu fused (commutes w/ shuffle)
  layer(t2, idx2, w2, b2, t3, 64, 128, 1);
  layer(t3, idx3, w3, b3, y3, 128, 12, 0);

  k_shuffle<<<(NB * 3 * 128 * 128 + 255) / 256, 256, 0, stream>>>(y3, out);
}